// HiPPO_39092792328888
// MI455X (gfx1250) — compile-verified
//
#include <hip/hip_runtime.h>

// ---------------------------------------------------------------------------
// HiPPO-LegS discretization + scan for gfx1250 (MI455X), wave32.
//
// Exploits: A lower-triangular (HiPPO-LegS), ALPHA=0.5 =>
//   M  = (I - h A)^{-1}  (lower triangular, h = 0.5/(k+1))
//   Ad = 2M - I
//   Bd = ss * M @ B
// Blocked triangular inversion over 16x16 tiles; tile products use
// V_WMMA_F32_16X16X4_F32 (fp32 WMMA: kernel is HBM-write bound, keep f32).
// ---------------------------------------------------------------------------

typedef __attribute__((ext_vector_type(2))) float v2f;
typedef __attribute__((ext_vector_type(8))) float v8f;

constexpr int kN  = 256;   // state dim
constexpr int kL  = 1024;  // sequence length
constexpr int kSX = 257;   // padded LDS row stride (257 % 64 == 1 -> conflict-free cols)

// --- WMMA fragment helpers (f32, 16x16x16 via four 16x16x4 steps) ----------
// A 16x4 layout : lanes 0-15 hold (M=lane, K=0..1 in v0/v1); lanes 16-31 K=2..3
// B 4x16 layout : lanes 0-15 hold (K=0/1 rows, N=lane); lanes 16-31 K=2/3
// C/D 16x16     : vgpr r -> M = r + 8*(lane>=16), N = lane & 15
__device__ __forceinline__ v8f tile_mma_16(const float* __restrict__ Abase, int astride,
                                           const float* __restrict__ Bbase, int bstride,
                                           v8f acc, int lane) {
  const int half = lane >> 4;
  const int l15  = lane & 15;
#pragma unroll
  for (int c = 0; c < 4; ++c) {
    v2f af, bf;
    const int ka = 4 * c + 2 * half;
    af.x = Abase[l15 * astride + ka + 0];
    af.y = Abase[l15 * astride + ka + 1];
    bf.x = Bbase[(ka + 0) * bstride + l15];
    bf.y = Bbase[(ka + 1) * bstride + l15];
    acc = __builtin_amdgcn_wmma_f32_16x16x4_f32(false, af, false, bf,
                                                (short)0, acc, false, false);
  }
  return acc;
}

__device__ __forceinline__ void tile_store_cd(float* __restrict__ dst, int stride,
                                              v8f d, int lane, float scale) {
  const int half = lane >> 4;
  const int l15  = lane & 15;
#pragma unroll
  for (int r = 0; r < 8; ++r)
    dst[(r + 8 * half) * stride + l15] = scale * d[r];
}

// ---------------------------------------------------------------------------
// Kernel 1: one workgroup per timestep k. Computes X = (I - h A)^{-1} in LDS,
// writes GBT_A = 2X - I (dense, zero above diagonal) and GBT_B = ss * X @ B.
// ---------------------------------------------------------------------------
__global__ __launch_bounds__(256)
void hippo_discretize(const float* __restrict__ A, const float* __restrict__ B,
                      float* __restrict__ GA, float* __restrict__ GB) {
  __shared__ float Xs[kN * kSX];      // 263,168 B : the inverse, padded rows
  __shared__ float Wsc[8 * 16 * 16];  //   8,192 B : per-wave 16x16 staging tile
  __shared__ float Bsh[kN];           //   1,024 B

  const int t    = threadIdx.x;
  const int w    = t >> 5;
  const int lane = t & 31;
  const int k    = blockIdx.x;
  const float ss = 1.0f / (float)(k + 1);
  const float h  = 0.5f * ss;         // ALPHA = 0.5

  // Zero X (upper triangle must read as 0 later), stage B.
  for (int i = t; i < kN * kSX; i += 256) Xs[i] = 0.0f;
  if (t < kN) Bsh[t] = B[t];
  __syncthreads();

  // --- Stage 1: invert the 16 diagonal 16x16 blocks (wave w -> blocks w, w+8).
  // Column n of inv(T_jj): x_n = 1/(1-h*a_nn);  x_m = h*(sum A[m][p] x_p)/(1-h*a_mm)
#pragma unroll 1
  for (int jj = 0; jj < 2; ++jj) {
    const int j = w + 8 * jj;
    if (lane < 16) {
      const int n = lane;
      const float* Ad = A  + (16 * j) * kN  + 16 * j;
      float*       Xd = Xs + (16 * j) * kSX + 16 * j;
      Xd[n * kSX + n] = 1.0f / (1.0f - h * Ad[n * kN + n]);
      for (int m = n + 1; m < 16; ++m) {
        float s = 0.0f;
        for (int p = n; p < m; ++p) s += Ad[m * kN + p] * Xd[p * kSX + n];
        Xd[m * kSX + n] = (h * s) / (1.0f - h * Ad[m * kN + m]);
      }
    }
  }
  __syncthreads();

  // --- Stage 2: distance-ordered off-diagonal sweep.
  // X_ij = h * X_ii * ( sum_{kk=j}^{i-1} A_ik X_kj ),  all 16x16x16 WMMA products.
  float* Wm = Wsc + w * 256;
#pragma unroll 1
  for (int d = 1; d < 16; ++d) {
#pragma unroll 1
    for (int p = w; p < 16 - d; p += 8) {          // wave-uniform => EXEC stays full
      const int j = p, i = p + d;
      v8f acc = {0.f, 0.f, 0.f, 0.f, 0.f, 0.f, 0.f, 0.f};
      for (int kk = j; kk < i; ++kk) {
        const float* Ab = A  + (16 * i)  * kN  + 16 * kk;   // L2-resident
        const float* Xb = Xs + (16 * kk) * kSX + 16 * j;
        acc = tile_mma_16(Ab, kN, Xb, kSX, acc, lane);
      }
      tile_store_cd(Wm, 16, acc, lane, 1.0f);      // D-layout -> LDS tile
      __builtin_amdgcn_wave_barrier();             // in-wave DS ops are in-order
      v8f acc2 = {0.f, 0.f, 0.f, 0.f, 0.f, 0.f, 0.f, 0.f};
      acc2 = tile_mma_16(Xs + (16 * i) * kSX + 16 * i, kSX, Wm, 16, acc2, lane);
      tile_store_cd(Xs + (16 * i) * kSX + 16 * j, kSX, acc2, lane, h);
    }
    __syncthreads();                               // level d feeds level d+1
  }

  // --- Stage 3: stream outputs. GBT_A[k] = 2X - I, fully coalesced float4 stores.
  float* GAk = GA + (size_t)k * kN * kN;
#pragma unroll 1
  for (int cc = 0; cc < 64; ++cc) {
    const int e   = cc * 1024 + t * 4;
    const int row = e >> 8;
    const int col = e & 255;
    const float* xr = Xs + row * kSX + col;
    float4 v;
    v.x = 2.0f * xr[0] - ((row == col + 0) ? 1.0f : 0.0f);
    v.y = 2.0f * xr[1] - ((row == col + 1) ? 1.0f : 0.0f);
    v.z = 2.0f * xr[2] - ((row == col + 2) ? 1.0f : 0.0f);
    v.w = 2.0f * xr[3] - ((row == col + 3) ? 1.0f : 0.0f);
    *(float4*)(GAk + e) = v;
  }
  // GBT_B[k][t] = ss * sum_j X[t][j] * B[j]  (stride 257 -> conflict-free lanes)
  float bacc = 0.0f;
  for (int j = 0; j <= t; ++j) bacc += Xs[t * kSX + j] * Bsh[j];
  GB[k * kN + t] = ss * bacc;
}

// ---------------------------------------------------------------------------
// Kernel 2: sequential scan (1024 dependent steps). One WG, 8 waves.
// Wave w owns rows [32w, 32w+32); each row is a coalesced 32-lane dot.
// ---------------------------------------------------------------------------
__global__ __launch_bounds__(256)
void hippo_scan(const float* __restrict__ f, const float* __restrict__ c0,
                const float* __restrict__ GA, const float* __restrict__ GB,
                float* __restrict__ c_all, float* __restrict__ y_all) {
  __shared__ float cvec[kN];
  __shared__ float nvec[kN];
  __shared__ float wsum[8];
  const int t = threadIdx.x, w = t >> 5, lane = t & 31;

  cvec[t] = c0[t];
  __syncthreads();

#pragma unroll 1
  for (int k = 0; k < kL; ++k) {
    const float fk = f[k];
    const float* Ak = GA + (size_t)k * kN * kN;
    const float* Bk = GB + k * kN;
#pragma unroll 1
    for (int r = 0; r < 32; ++r) {
      const int row = w * 32 + r;
      const float* arow = Ak + row * kN;
      float s = 0.0f;
#pragma unroll
      for (int c = 0; c < 8; ++c) {
        const int j = c * 32 + lane;
        s += arow[j] * cvec[j];
      }
#pragma unroll
      for (int off = 16; off > 0; off >>= 1) s += __shfl_xor(s, off, 32);
      if (lane == 0) nvec[row] = s + Bk[row] * fk;
    }
    __syncthreads();

    const float cv = nvec[t];
    c_all[k * kN + t] = cv;
    float s = cv;
#pragma unroll
    for (int off = 16; off > 0; off >>= 1) s += __shfl_xor(s, off, 32);
    if (lane == 0) wsum[w] = s;
    __syncthreads();
    if (t == 0) {
      float y = 0.0f;
#pragma unroll
      for (int i = 0; i < 8; ++i) y += wsum[i];
      y_all[k] = y;
    }
    cvec[t] = cv;
    __syncthreads();
  }
}

// ---------------------------------------------------------------------------
extern "C" void kernel_launch(void* const* d_in, const int* in_sizes, int n_in,
                              void* d_out, int out_size, void* d_ws, size_t ws_size,
                              hipStream_t stream) {
  (void)in_sizes; (void)n_in; (void)out_size; (void)d_ws; (void)ws_size;
  const float* f  = (const float*)d_in[0];   // (L,1)
  const float* c0 = (const float*)d_in[1];   // (N,1)
  const float* A  = (const float*)d_in[2];   // (N,N)
  const float* B  = (const float*)d_in[3];   // (N,1)

  float* out   = (float*)d_out;
  float* c_all = out;                               // L*N
  float* y_all = out + (size_t)kL * kN;             // L
  float* GA    = y_all + kL;                        // L*N*N
  float* GB    = GA + (size_t)kL * kN * kN;         // L*N

  hippo_discretize<<<kL, 256, 0, stream>>>(A, B, GA, GB);
  hippo_scan<<<1, 256, 0, stream>>>(f, c0, GA, GB, c_all, y_all);
}